// StemGNNBlock_43353399886132
// MI455X (gfx1250) — compile-verified
//
#include <hip/hip_runtime.h>
#include <math.h>

// ---------------- sizes ----------------
#define B_  8
#define N_  256
#define T_  96
#define F_  128
#define H_  8
#define HD_ 16
#define KF  49      // rfft bins for T=96
#define KP  64      // padded bin count (multiple of 16)
#define PI_ 3.14159265358979323846f

typedef float v2f __attribute__((ext_vector_type(2)));
typedef float v8f __attribute__((ext_vector_type(8)));
typedef int   v4i_ __attribute__((vector_size(16)));

// ---------------- async global->LDS copy (gfx1250) with fallback ----------------
#if defined(__has_builtin)
#if __has_builtin(__builtin_amdgcn_global_load_async_to_lds_b128) && \
    __has_builtin(__builtin_amdgcn_s_wait_asynccnt)
#define HAVE_ASYNC_LDS 1
#endif
#endif

#define AS1 __attribute__((address_space(1)))
#define AS3 __attribute__((address_space(3)))

// copy 16 bytes (4 floats) from global to LDS, per-lane
__device__ __forceinline__ void cp16(float* lds_dst, const float* g_src) {
#ifdef HAVE_ASYNC_LDS
    __builtin_amdgcn_global_load_async_to_lds_b128(
        (AS1 v4i_*)(g_src), (AS3 v4i_*)(lds_dst), 0, 0);
#else
    *(float4*)lds_dst = *(const float4*)g_src;
#endif
}
__device__ __forceinline__ void cp_wait() {
#ifdef HAVE_ASYNC_LDS
    __builtin_amdgcn_s_wait_asynccnt(0);
#endif
}

// ---------------- WMMA f32 16x16x4 helpers (CDNA5 wave32) ----------------
// A (16x4): lane l: m=l&15, half=l>>4 ; V0->K=2h, V1->K=2h+1   (ISA 7.12.2)
// B (4x16): lane l: n=l&15, half=l>>4 ; V0->K=2h row, V1->K=2h+1 row
// C/D (16x16): VGPR r: lanes0-15 -> M=r, lanes16-31 -> M=r+8
__device__ __forceinline__ v8f zero8() {
    v8f z = {0.f,0.f,0.f,0.f,0.f,0.f,0.f,0.f};
    return z;
}
__device__ __forceinline__ v2f ldA(const float* A, int lda, int lane) {
    int m = lane & 15, h = lane >> 4;
    v2f a;
    a.x = A[m * lda + 2 * h];
    a.y = A[m * lda + 2 * h + 1];
    return a;
}
__device__ __forceinline__ v2f ldB(const float* Bp, int ldb, int lane) {
    int n = lane & 15, h = lane >> 4;
    v2f b;
    b.x = Bp[(2 * h) * ldb + n];
    b.y = Bp[(2 * h + 1) * ldb + n];
    return b;
}
__device__ __forceinline__ v8f wmma4(v2f a, v2f b, v8f c) {
    return __builtin_amdgcn_wmma_f32_16x16x4_f32(
        /*neg_a=*/false, a, /*neg_b=*/false, b,
        /*c_mod=*/(short)0, c, /*reuse_a=*/false, /*reuse_b=*/false);
}
__device__ __forceinline__ void stC(float* C, int ldc, int lane, v8f acc) {
    int n = lane & 15, h = lane >> 4;
#pragma unroll
    for (int r = 0; r < 8; ++r) C[(r + 8 * h) * ldc + n] = acc[r];
}

// ---------------- kernel 0: DFT / IDFT coefficient tables ----------------
// tabs layout: Wr[64*96] Wi[64*96] Ar[96*64] Ai[96*64]
__global__ void k_tables(float* tabs) {
    float* Wr = tabs;
    float* Wi = tabs + KP * T_;
    float* Ar = Wi + KP * T_;
    float* Ai = Ar + T_ * KP;
    int tid = blockIdx.x * blockDim.x + threadIdx.x;
    int stride = gridDim.x * blockDim.x;
    const float w = 2.0f * PI_ / (float)T_;
    for (int idx = tid; idx < KP * T_; idx += stride) {
        int k = idx / T_, t = idx % T_;
        float vr = 0.f, vi = 0.f;
        if (k < KF) {
            float th = w * (float)k * (float)t;
            vr = cosf(th);
            vi = -sinf(th);
        }
        Wr[idx] = vr;
        Wi[idx] = vi;
    }
    for (int idx = tid; idx < T_ * KP; idx += stride) {
        int t = idx / KP, k = idx % KP;
        float ar = 0.f, ai = 0.f;
        if (k < KF) {
            float c = (k == 0 || k == 48) ? 1.0f : 2.0f;
            float th = w * (float)k * (float)t;
            ar = c * cosf(th) / (float)T_;
            ai = (k == 0 || k == 48) ? 0.0f : (-c * sinf(th) / (float)T_);
        }
        Ar[idx] = ar;
        Ai[idx] = ai;
    }
}

// ---------------- kernel 1: spectral block (per (b,n)) ----------------
// DFT(WMMA) -> grouped freq conv(WMMA,3 taps)+bias+GELU -> IDFT(WMMA) -> LN + T-mean
__global__ void __launch_bounds__(256) k_spectral(
    const float* __restrict__ x, const float* __restrict__ spec_w,
    const float* __restrict__ spec_b, const float* __restrict__ ln_g,
    const float* __restrict__ ln_b, const float* __restrict__ tabs,
    float* __restrict__ x_sp, float* __restrict__ x_mean)
{
    __shared__ float lds[37376];              // 146 KB (WGP has 320 KB LDS)
    float* xt = lds;                          // [96][128]  (reused as Yr [64][128])
    float* Xr = lds + 12288;                  // [66][128]  row0 & row65 zero pad
    float* Xi = Xr + 66 * 128;                // [66][128]
    float* Yr = xt;                           // [64][128]
    float* Yi = Xi + 66 * 128;                // [64][128]
    float* ot = Xr;                           // [96][128]  (irfft output, overlays Xr/Xi)

    const float* Wr = tabs;
    const float* Wi = tabs + KP * T_;
    const float* Ar = Wi + KP * T_;
    const float* Ai = Ar + T_ * KP;

    int bn = blockIdx.x;
    int tid = threadIdx.x, wave = tid >> 5, lane = tid & 31;
    const float* xin = x + (size_t)bn * (T_ * F_);

    for (int i = tid * 4; i < T_ * F_; i += 1024)
        cp16(xt + i, xin + i);
    for (int i = tid; i < 128; i += 256) {
        Xr[i] = 0.f; Xi[i] = 0.f;
        Xr[65 * 128 + i] = 0.f; Xi[65 * 128 + i] = 0.f;
    }
    cp_wait();
    __syncthreads();

    // ---- DFT: X(64x128) = W(64x96) @ xt(96x128) ----
    for (int tt = wave; tt < 64; tt += 8) {
        int ri = tt >> 5, mt = (tt >> 3) & 3, nt = tt & 7;
        const float* W = ri ? Wi : Wr;
        float* dst = ri ? Xi : Xr;
        v8f acc = zero8();
        for (int k0 = 0; k0 < 96; k0 += 4) {
            v2f a = ldA(W + (mt * 16) * 96 + k0, 96, lane);
            v2f b = ldB(xt + k0 * 128 + nt * 16, 128, lane);
            acc = wmma4(a, b, acc);
        }
        stC(dst + (1 + mt * 16) * 128 + nt * 16, 128, lane, acc);
    }
    __syncthreads();

    // ---- grouped conv over freq (kernel 3, pad 1) + bias + exact GELU ----
    for (int tt = wave; tt < 64; tt += 8) {
        int ri = tt >> 5, mt = (tt >> 3) & 3, nt = tt & 7;
        const float* src = ri ? Xi : Xr;
        float* dst = ri ? Yi : Yr;
        const float* Wc = spec_w + (ri ? (size_t)128 * 384 : 0);
        int n = lane & 15, h = lane >> 4;
        int o = nt * 16 + n;
        v8f acc = zero8();
        for (int dk = 0; dk < 3; ++dk) {
            const float* Abase = src + (mt * 16 + dk) * 128; // row p+dk (buffer has -1 pad)
            for (int k0 = 0; k0 < 128; k0 += 4) {
                v2f a = ldA(Abase + k0, 128, lane);
                v2f b;
                b.x = Wc[(size_t)o * 384 + (k0 + 2 * h) * 3 + dk];
                b.y = Wc[(size_t)o * 384 + (k0 + 2 * h + 1) * 3 + dk];
                acc = wmma4(a, b, acc);
            }
        }
        float bias = spec_b[(ri ? 128 : 0) + o];
#pragma unroll
        for (int r = 0; r < 8; ++r) {
            float v = acc[r] + bias;
            v = 0.5f * v * (1.0f + erff(v * 0.70710678118f)); // exact GELU
            dst[(mt * 16 + r + 8 * h) * 128 + o] = v;
        }
    }
    __syncthreads();

    // ---- IDFT: ot(96x128) = Ar(96x64)@Yr + Ai(96x64)@Yi ----
    for (int tt = wave; tt < 48; tt += 8) {
        int mt = tt / 8, nt = tt % 8;
        v8f acc = zero8();
        for (int k0 = 0; k0 < 64; k0 += 4) {
            v2f a = ldA(Ar + (mt * 16) * 64 + k0, 64, lane);
            v2f b = ldB(Yr + k0 * 128 + nt * 16, 128, lane);
            acc = wmma4(a, b, acc);
            a = ldA(Ai + (mt * 16) * 64 + k0, 64, lane);
            b = ldB(Yi + k0 * 128 + nt * 16, 128, lane);
            acc = wmma4(a, b, acc);
        }
        stC(ot + (mt * 16) * 128 + nt * 16, 128, lane, acc);
    }
    __syncthreads();

    // ---- LayerNorm over F per t-row, then mean over T ----
    float* outg = x_sp + (size_t)bn * (T_ * F_);
    for (int t = wave; t < 96; t += 8) {
        float* row = ot + t * 128;
        float s = 0.f, s2 = 0.f;
        for (int f = lane; f < 128; f += 32) { float v = row[f]; s += v; s2 += v * v; }
        for (int off = 16; off; off >>= 1) {
            s += __shfl_xor(s, off, 32);
            s2 += __shfl_xor(s2, off, 32);
        }
        float mu = s / 128.f;
        float rstd = rsqrtf(s2 / 128.f - mu * mu + 1e-5f);
        for (int f = lane; f < 128; f += 32) {
            float v = (row[f] - mu) * rstd * ln_g[f] + ln_b[f];
            outg[t * 128 + f] = v;
            row[f] = v;
        }
    }
    __syncthreads();
    for (int f = tid; f < 128; f += 256) {
        float s = 0.f;
        for (int t = 0; t < 96; ++t) s += ot[t * 128 + f];
        x_mean[(size_t)bn * 128 + f] = s / 96.f;
    }
}

// ---------------- kernel 2: q/k projections (2048x128 @ 128x128) ----------------
__global__ void __launch_bounds__(256) k_qk(
    const float* __restrict__ x_mean,
    const float* __restrict__ q_w, const float* __restrict__ q_b,
    const float* __restrict__ k_w, const float* __restrict__ k_b,
    float* __restrict__ q, float* __restrict__ k)
{
    __shared__ float sA[64 * 128];
    int m0 = blockIdx.x * 64;
    int tid = threadIdx.x, wave = tid >> 5, lane = tid & 31;
    for (int i = tid * 4; i < 64 * 128; i += 1024)
        cp16(sA + i, x_mean + (size_t)m0 * 128 + i);
    cp_wait();
    __syncthreads();
    for (int tt = wave; tt < 64; tt += 8) {
        int qk = tt >> 5, mt = (tt >> 3) & 3, nt = tt & 7;
        const float* W = qk ? k_w : q_w;
        const float* bias = qk ? k_b : q_b;
        float* dst = qk ? k : q;
        int n = lane & 15, h = lane >> 4;
        int o = nt * 16 + n;
        v8f acc = zero8();
        for (int k0 = 0; k0 < 128; k0 += 4) {
            v2f a = ldA(sA + mt * 16 * 128 + k0, 128, lane);
            v2f b;
            b.x = W[(size_t)o * 128 + k0 + 2 * h];
            b.y = W[(size_t)o * 128 + k0 + 2 * h + 1];
            acc = wmma4(a, b, acc);
        }
        float bb = bias[o];
#pragma unroll
        for (int r = 0; r < 8; ++r)
            dst[(size_t)(m0 + mt * 16 + r + 8 * h) * 128 + o] = acc[r] + bb;
    }
}

// ---------------- kernel 3: attention scores per (b,h): 256x256, K=16 ----------------
__global__ void __launch_bounds__(256) k_scores(
    const float* __restrict__ q, const float* __restrict__ k,
    const float* __restrict__ temperature, float* __restrict__ S)
{
    __shared__ float sQ[256 * 16];   // [n][d]
    __shared__ float sKT[16 * 256];  // [d][m]
    int b = blockIdx.x >> 3, h = blockIdx.x & 7;
    int tid = threadIdx.x, wave = tid >> 5, lane = tid & 31;
    for (int i = tid; i < 256 * 16; i += 256) {
        int n = i >> 4, d = i & 15;
        float qv = q[(size_t)(b * 256 + n) * 128 + h * 16 + d];
        float kv = k[(size_t)(b * 256 + n) * 128 + h * 16 + d];
        sQ[i] = qv;
        sKT[d * 256 + n] = kv;
    }
    __syncthreads();
    float invT = 1.0f / temperature[0];
    float* Sb = S + (size_t)blockIdx.x * 256 * 256;
    for (int tt = wave; tt < 256; tt += 8) {
        int mt = tt >> 4, nt = tt & 15;
        v8f acc = zero8();
        for (int k0 = 0; k0 < 16; k0 += 4) {
            v2f a = ldA(sQ + mt * 16 * 16 + k0, 16, lane);
            v2f bf = ldB(sKT + k0 * 256 + nt * 16, 256, lane);
            acc = wmma4(a, bf, acc);
        }
        int n = lane & 15, hh = lane >> 4;
#pragma unroll
        for (int r = 0; r < 8; ++r)
            Sb[(size_t)(mt * 16 + r + 8 * hh) * 256 + nt * 16 + n] = acc[r] * invT;
    }
}

// ---------------- kernel 4: softmax over m + mean over heads -> adj ----------------
__global__ void __launch_bounds__(256) k_softmax(const float* __restrict__ S,
                                                 float* __restrict__ adj)
{
    __shared__ float red[256];
    int b = blockIdx.x >> 8, n = blockIdx.x & 255;
    int tid = threadIdx.x;
    float acc = 0.f;
    for (int h = 0; h < 8; ++h) {
        const float* row = S + (((size_t)(b * 8 + h)) * 256 + n) * 256;
        float v = row[tid];
        red[tid] = v; __syncthreads();
        for (int off = 128; off; off >>= 1) {
            if (tid < off) red[tid] = fmaxf(red[tid], red[tid + off]);
            __syncthreads();
        }
        float mx = red[0]; __syncthreads();
        float e = __expf(v - mx);
        red[tid] = e; __syncthreads();
        for (int off = 128; off; off >>= 1) {
            if (tid < off) red[tid] += red[tid + off];
            __syncthreads();
        }
        float sum = red[0]; __syncthreads();
        acc += e / sum;
    }
    adj[(size_t)blockIdx.x * 256 + tid] = acc * 0.125f;
}

// ---------------- kernel 5: attended = adj @ x_sp, residual + attn LN (per (b,t)) ----
__global__ void __launch_bounds__(256) k_attended(
    const float* __restrict__ adj, const float* __restrict__ x_sp,
    const float* __restrict__ ln_g, const float* __restrict__ ln_b,
    float* __restrict__ x_g)
{
    __shared__ float sB[256 * 128];           // x_sp[b, :, t, :]  (128 KB)
    int t = blockIdx.x % 96;
    int b = blockIdx.x / 96;
    int tid = threadIdx.x, wave = tid >> 5, lane = tid & 31;
    for (int m = wave; m < 256; m += 8) {
        const float* src = x_sp + ((size_t)(b * 256 + m) * 96 + t) * 128;
        cp16(sB + m * 128 + lane * 4, src + lane * 4);
    }
    cp_wait();
    __syncthreads();

    for (int mt = wave * 2; mt < wave * 2 + 2; ++mt) {
        v8f acc[8];
#pragma unroll
        for (int nt = 0; nt < 8; ++nt) acc[nt] = zero8();
        const float* Ab = adj + ((size_t)b * 256 + mt * 16) * 256;
        for (int k0 = 0; k0 < 256; k0 += 4) {
            v2f a = ldA(Ab + k0, 256, lane);
#pragma unroll
            for (int nt = 0; nt < 8; ++nt) {
                v2f bf = ldB(sB + k0 * 128 + nt * 16, 128, lane);
                acc[nt] = wmma4(a, bf, acc[nt]);
            }
        }
        int n = lane & 15, h = lane >> 4;
#pragma unroll
        for (int r = 0; r < 8; ++r) {
            int row = r + 8 * h;
            int nrow = mt * 16 + row;
            float vals[8];
            float s = 0.f, s2 = 0.f;
#pragma unroll
            for (int nt = 0; nt < 8; ++nt) {
                float v = acc[nt][r] + sB[nrow * 128 + nt * 16 + n]; // + x_sp residual
                vals[nt] = v; s += v; s2 += v * v;
            }
            // reduce across the 16 lanes of this half (xor masks stay in-half)
            for (int off = 1; off < 16; off <<= 1) {
                s += __shfl_xor(s, off, 32);
                s2 += __shfl_xor(s2, off, 32);
            }
            float mu = s / 128.f;
            float rstd = rsqrtf(s2 / 128.f - mu * mu + 1e-5f);
            float* dst = x_g + ((size_t)(b * 256 + nrow) * 96 + t) * 128;
#pragma unroll
            for (int nt = 0; nt < 8; ++nt) {
                int f = nt * 16 + n;
                dst[f] = (vals[nt] - mu) * rstd * ln_g[f] + ln_b[f];
            }
        }
    }
}

// ---------------- kernel 6: temporal depthwise conv + GLU + residual + out LN ----------
__global__ void __launch_bounds__(256) k_glu(
    const float* __restrict__ x, const float* __restrict__ x_g,
    const float* __restrict__ temp_w, const float* __restrict__ temp_b,
    const float* __restrict__ glu_w, const float* __restrict__ glu_b,
    const float* __restrict__ ln_g, const float* __restrict__ ln_b,
    float* __restrict__ out)
{
    __shared__ float lds[36864];              // 144 KB
    float* xg = lds;                          // [96][128] (dead after conv; reused as gleft)
    float* ct = lds + 12288;                  // [96][128] conv output (GEMM A)
    float* gleft = lds;                       // g[..., 0:128]
    float* gright = lds + 24576;              // g[..., 128:256]

    int bn = blockIdx.x;
    int tid = threadIdx.x, wave = tid >> 5, lane = tid & 31;
    const float* src = x_g + (size_t)bn * (T_ * F_);
    for (int i = tid * 4; i < T_ * F_; i += 1024)
        cp16(xg + i, src + i);
    cp_wait();
    __syncthreads();
    for (int i = tid; i < T_ * F_; i += 256) {
        int t = i >> 7, f = i & 127;
        float v = temp_w[f * 3 + 1] * xg[i] + temp_b[f];
        if (t > 0)  v += temp_w[f * 3 + 0] * xg[i - 128];
        if (t < 95) v += temp_w[f * 3 + 2] * xg[i + 128];
        ct[i] = v;
    }
    __syncthreads();

    // g(96x256) = ct(96x128) @ glu_w^T(128x256)
    for (int tt = wave; tt < 96; tt += 8) {
        int mt = tt / 16, nt = tt % 16;
        int n = lane & 15, h = lane >> 4;
        int o = nt * 16 + n;
        v8f acc = zero8();
        for (int k0 = 0; k0 < 128; k0 += 4) {
            v2f a = ldA(ct + mt * 16 * 128 + k0, 128, lane);
            v2f bf;
            bf.x = glu_w[(size_t)o * 128 + k0 + 2 * h];
            bf.y = glu_w[(size_t)o * 128 + k0 + 2 * h + 1];
            acc = wmma4(a, bf, acc);
        }
        float bb = glu_b[o];
        float* dbase = (nt < 8) ? gleft : gright;
        int oc = (nt < 8) ? o : (o - 128);
#pragma unroll
        for (int r = 0; r < 8; ++r)
            dbase[(mt * 16 + r + 8 * h) * 128 + oc] = acc[r] + bb;
    }
    __syncthreads();

    // GLU + residual with original x + final LayerNorm
    const float* xin = x + (size_t)bn * (T_ * F_);
    float* dst = out + (size_t)bn * (T_ * F_);
    for (int t = wave; t < 96; t += 8) {
        float vals[4];
        float s = 0.f, s2 = 0.f;
#pragma unroll
        for (int j = 0; j < 4; ++j) {
            int f = lane + 32 * j;
            float l = gleft[t * 128 + f], rr = gright[t * 128 + f];
            float v = xin[t * 128 + f] + l * (1.0f / (1.0f + __expf(-rr)));
            vals[j] = v; s += v; s2 += v * v;
        }
        for (int off = 16; off; off >>= 1) {
            s += __shfl_xor(s, off, 32);
            s2 += __shfl_xor(s2, off, 32);
        }
        float mu = s / 128.f;
        float rstd = rsqrtf(s2 / 128.f - mu * mu + 1e-5f);
#pragma unroll
        for (int j = 0; j < 4; ++j) {
            int f = lane + 32 * j;
            dst[t * 128 + f] = (vals[j] - mu) * rstd * ln_g[f] + ln_b[f];
        }
    }
}

// ---------------- launcher ----------------
extern "C" void kernel_launch(void* const* d_in, const int* in_sizes, int n_in,
                              void* d_out, int out_size, void* d_ws, size_t ws_size,
                              hipStream_t stream) {
    (void)in_sizes; (void)n_in; (void)out_size; (void)ws_size;
    const float* x         = (const float*)d_in[0];
    const float* spec_w    = (const float*)d_in[1];
    const float* spec_b    = (const float*)d_in[2];
    const float* spec_ln_g = (const float*)d_in[3];
    const float* spec_ln_b = (const float*)d_in[4];
    const float* q_w       = (const float*)d_in[5];
    const float* q_b       = (const float*)d_in[6];
    const float* k_w       = (const float*)d_in[7];
    const float* k_b       = (const float*)d_in[8];
    const float* temp      = (const float*)d_in[9];
    const float* attn_ln_g = (const float*)d_in[10];
    const float* attn_ln_b = (const float*)d_in[11];
    const float* temp_w    = (const float*)d_in[12];
    const float* temp_b    = (const float*)d_in[13];
    const float* glu_w     = (const float*)d_in[14];
    const float* glu_b     = (const float*)d_in[15];
    const float* out_ln_g  = (const float*)d_in[16];
    const float* out_ln_b  = (const float*)d_in[17];

    const size_t NOUT = (size_t)B_ * N_ * T_ * F_;   // 25165824
    float* out = (float*)d_out;
    float* adj = out + NOUT;                          // (B,N,N) tail of d_out

    float* ws     = (float*)d_ws;
    float* tabs   = ws;                               // 24576
    float* x_sp   = tabs + 24576;                     // 25165824
    float* x_mean = x_sp + NOUT;                      // 262144
    float* q      = x_mean + (size_t)B_ * N_ * F_;    // 262144
    float* k      = q + (size_t)B_ * N_ * F_;         // 262144
    float* S      = k + (size_t)B_ * N_ * F_;         // 4194304
    float* x_g    = S + (size_t)B_ * H_ * N_ * N_;    // 25165824

    k_tables  <<<32, 256, 0, stream>>>(tabs);
    k_spectral<<<B_ * N_, 256, 0, stream>>>(x, spec_w, spec_b, spec_ln_g, spec_ln_b,
                                            tabs, x_sp, x_mean);
    k_qk      <<<32, 256, 0, stream>>>(x_mean, q_w, q_b, k_w, k_b, q, k);
    k_scores  <<<B_ * H_, 256, 0, stream>>>(q, k, temp, S);
    k_softmax <<<B_ * N_, 256, 0, stream>>>(S, adj);
    k_attended<<<B_ * T_, 256, 0, stream>>>(adj, x_sp, attn_ln_g, attn_ln_b, x_g);
    k_glu     <<<B_ * N_, 256, 0, stream>>>(x, x_g, temp_w, temp_b, glu_w, glu_b,
                                            out_ln_g, out_ln_b, out);
}